// MambaStateSpace_55722905698421
// MI455X (gfx1250) — compile-verified
//
#include <hip/hip_runtime.h>
#include <math.h>

// ---------------------------------------------------------------------------
// Problem constants (from reference)
// ---------------------------------------------------------------------------
#define INPUT_DIM 512
#define D_STATE   16
#define BATCH     32
#define SEQ       8192

typedef float v2f __attribute__((ext_vector_type(2)));
typedef float v8f __attribute__((ext_vector_type(8)));

#if __has_builtin(__builtin_amdgcn_tanhf)
#define TANH_F32(x) __builtin_amdgcn_tanhf(x)
#else
#define TANH_F32(x) tanhf(x)
#endif

// ds_swizzle SWAPX16 on wave32 (full-EXEC only!): xor_mask=0x10, and_mask=0x1f
__device__ __forceinline__ float swap_half(float v) {
    return __int_as_float(__builtin_amdgcn_ds_swizzle(__float_as_int(v), 0x401F));
}

// ---------------------------------------------------------------------------
// Kernel 1: fold W_proj^T @ B into a single 512-vector; b_eff = b_proj . B
//   ws[0..511]  = w_eff
//   ws[512]     = b_eff
// ---------------------------------------------------------------------------
__global__ void fold_kernel(const float* __restrict__ Wp,
                            const float* __restrict__ Bv,
                            const float* __restrict__ bproj,
                            float* __restrict__ ws) {
    const int i = threadIdx.x;   // 512 threads
    float acc = 0.f;
#pragma unroll
    for (int d = 0; d < D_STATE; ++d)
        acc += Wp[d * INPUT_DIM + i] * Bv[d];
    ws[i] = acc;
    if (i == 0) {
        float be = 0.f;
#pragma unroll
        for (int d = 0; d < D_STATE; ++d) be += bproj[d] * Bv[d];
        ws[INPUT_DIM] = be;
    }
}

// ---------------------------------------------------------------------------
// Kernel 2: u[b,t] = x[b,t,:] . w_eff + b_eff    (memory-bound GEMV, 512 MiB)
// One wave per token: 32 lanes x 4 float4 = 512 contiguous floats, fully
// coalesced; wave32 xor-shuffle reduction; lane 0 writes u.
// HBM floor: 512 MiB / 23.3 TB/s ~ 23 us; compute (0.27 GFLOP) is noise.
// ---------------------------------------------------------------------------
__global__ void gemv_kernel(const float* __restrict__ x,
                            const float* __restrict__ ws,
                            float* __restrict__ u) {
    const int lane          = threadIdx.x & 31;
    const int wave_in_blk   = threadIdx.x >> 5;
    const int waves_per_blk = blockDim.x >> 5;
    const int wave          = blockIdx.x * waves_per_blk + wave_in_blk;
    const int nwaves        = gridDim.x * waves_per_blk;

    const float4* x4 = (const float4*)x;
    const float   b_eff = ws[INPUT_DIM];

    // each lane's 4 chunks of w_eff, resident in registers
    float4 w[4];
    const float4* w4 = (const float4*)ws;
#pragma unroll
    for (int i = 0; i < 4; ++i) w[i] = w4[i * 32 + lane];

    const int ntok = BATCH * SEQ;
    for (int tok = wave; tok < ntok; tok += nwaves) {
        const float4* row = x4 + (size_t)tok * (INPUT_DIM / 4);
        // prefetch the next row this wave will touch (global_prefetch_b8)
        if (tok + nwaves < ntok) {
            const float4* nrow = x4 + (size_t)(tok + nwaves) * (INPUT_DIM / 4);
#pragma unroll
            for (int i = 0; i < 4; ++i)
                __builtin_prefetch((const void*)(nrow + i * 32 + lane), 0, 0);
        }
        float acc = 0.f;
#pragma unroll
        for (int i = 0; i < 4; ++i) {
            float4 v = row[i * 32 + lane];
            acc += v.x * w[i].x + v.y * w[i].y + v.z * w[i].z + v.w * w[i].w;
        }
#pragma unroll
        for (int off = 16; off > 0; off >>= 1)
            acc += __shfl_xor(acc, off, 32);
        if (lane == 0) u[tok] = acc + b_eff;   // layout: u[b*SEQ + t]
    }
}

// ---------------------------------------------------------------------------
// Kernel 3: sequential scan  s_t = tanh(A @ s_{t-1}^T + u_t), 8192 steps.
// One wave handles 16 batches: D(16x16) = A(16x16) x S^T(16x16) via 4 chained
// V_WMMA_F32_16X16X4_F32 (K=16). Batches live in the N dimension, so columns
// stay pinned to lanes across steps.
//
// D -> next-B reshape is done as 4 register-pair half swaps:
//   pair (X=ns[a], Y=ns[b]) needs outputs {X.lo | Y.lo} and {X.hi | Y.hi}.
//   q = hi ? X : Y              (local cndmask)
//   w = swap_half(q)            (ONE unconditional full-EXEC ds_swizzle)
//   out0 = hi ? w : X ; out1 = hi ? Y : w
// -> 4 swizzles + 12 cndmask per step, zero exec-mask manipulation.
//
// WMMA f32 16x16x4 layouts (ISA 7.12.2):
//   A (16x4):  lane m=l%16; vgpr r holds K = 2*(l/16) + r         -> v2f
//   B (4x16):  lane n=l%16; vgpr r holds K = 2*(l/16) + r         -> v2f
//   D (16x16): lane n=l%16; vgpr r holds M = r + 8*(l/16)         -> v8f
// ---------------------------------------------------------------------------
__global__ __launch_bounds__(32, 1) void scan_kernel(
    const float* __restrict__ A,
    const float* __restrict__ C,
    const float* __restrict__ Dm,
    const float* __restrict__ u,
    float* __restrict__ out) {

    const int lane = threadIdx.x & 31;
    const int col  = lane & 15;   // N index == batch-within-group
    const int hi   = lane >> 4;   // lane half
    const int b    = blockIdx.x * 16 + col;

    // A in WMMA-A layout, 4 K-chunks of width 4 (m = col)
    v2f a[4];
#pragma unroll
    for (int c = 0; c < 4; ++c) {
        a[c].x = A[col * D_STATE + 4 * c + 2 * hi + 0];
        a[c].y = A[col * D_STATE + 4 * c + 2 * hi + 1];
    }
    // C coefficients for the 8 rows this lane's D registers hold
    float cc[8];
#pragma unroll
    for (int r = 0; r < 8; ++r) cc[r] = C[r + 8 * hi];
    const float dsc = Dm[0];

    // state S^T in WMMA-B layout: sb[c] holds K = 4c+2*hi+{0,1}, N = col
    v2f sb[4];
#pragma unroll
    for (int c = 0; c < 4; ++c) { sb[c].x = 0.f; sb[c].y = 0.f; }

    const float4* urow = (const float4*)(u + (size_t)b * SEQ);
    float accS = 0.f, accU = 0.f;

    // 2-deep register pipeline on u: load runs 8 steps ahead of use.
    // (u region has slack floats in the workspace, so the tail over-read of
    //  up to 2 float4 groups is safe and the loop stays branch-free.)
    float4 ucur = urow[0];
    float4 unxt = urow[1];
    for (int t4 = 0; t4 < SEQ / 4; ++t4) {
        float4 u2 = urow[t4 + 2];           // unconditional, slack-covered
        float uv[4] = {ucur.x, ucur.y, ucur.z, ucur.w};
#pragma unroll
        for (int k = 0; k < 4; ++k) {
            v8f d = {0.f, 0.f, 0.f, 0.f, 0.f, 0.f, 0.f, 0.f};
            d = __builtin_amdgcn_wmma_f32_16x16x4_f32(false, a[0], false, sb[0], (short)0, d, false, false);
            d = __builtin_amdgcn_wmma_f32_16x16x4_f32(false, a[1], false, sb[1], (short)0, d, false, false);
            d = __builtin_amdgcn_wmma_f32_16x16x4_f32(false, a[2], false, sb[2], (short)0, d, false, false);
            d = __builtin_amdgcn_wmma_f32_16x16x4_f32(false, a[3], false, sb[3], (short)0, d, false, false);

            const float uval = uv[k];
            float ns[8];
#pragma unroll
            for (int r = 0; r < 8; ++r) ns[r] = TANH_F32(d[r] + uval);

            // output accumulation: y_t = ns . C + u_t * D
#pragma unroll
            for (int r = 0; r < 8; ++r) accS += ns[r] * cc[r];
            accU += uval;

            // --- reshape D-layout ns -> B-layout sb (4 pair half-swaps) ---
            // ns[r] at lane l == s[b=l%16][d = r + 8*(l/16)]
            // pairs: (0,2) (1,3) (4,6) (5,7)
            float q0 = hi ? ns[0] : ns[2];
            float q1 = hi ? ns[1] : ns[3];
            float q2 = hi ? ns[4] : ns[6];
            float q3 = hi ? ns[5] : ns[7];
            float w0 = swap_half(q0);       // full-EXEC swizzles
            float w1 = swap_half(q1);
            float w2 = swap_half(q2);
            float w3 = swap_half(q3);
            sb[0].x = hi ? w0 : ns[0];      // {s0 | s2}
            sb[2].x = hi ? ns[2] : w0;      // {s8 | s10}
            sb[0].y = hi ? w1 : ns[1];      // {s1 | s3}
            sb[2].y = hi ? ns[3] : w1;      // {s9 | s11}
            sb[1].x = hi ? w2 : ns[4];      // {s4 | s6}
            sb[3].x = hi ? ns[6] : w2;      // {s12| s14}
            sb[1].y = hi ? w3 : ns[5];      // {s5 | s7}
            sb[3].y = hi ? ns[7] : w3;      // {s13| s15}
        }
        ucur = unxt;
        unxt = u2;
    }

    // combine the two lane-halves (rows 0-7 and 8-15) for each batch column
    float tot = accS + swap_half(accS);
    if (hi == 0) out[b] = (tot + accU * dsc) / (float)SEQ;
}

// ---------------------------------------------------------------------------
// Launch
//   d_in: x, A, B, C, D, W_proj, b_proj   (all f32)
//   d_out: 32 floats
//   d_ws layout (floats): [0..511] w_eff, [512] b_eff,
//                         [640 .. 640+262144+16) u  (+16 floats tail slack
//                         so the scan's branch-free 2-deep prefetch may
//                         over-read harmlessly)
// ---------------------------------------------------------------------------
extern "C" void kernel_launch(void* const* d_in, const int* in_sizes, int n_in,
                              void* d_out, int out_size, void* d_ws, size_t ws_size,
                              hipStream_t stream) {
    const float* x     = (const float*)d_in[0];
    const float* A     = (const float*)d_in[1];
    const float* Bv    = (const float*)d_in[2];
    const float* C     = (const float*)d_in[3];
    const float* Dm    = (const float*)d_in[4];
    const float* Wp    = (const float*)d_in[5];
    const float* bproj = (const float*)d_in[6];

    float* ws = (float*)d_ws;
    float* u  = ws + 640;                 // 16B-aligned region for u[b*SEQ+t]

    fold_kernel<<<1, INPUT_DIM, 0, stream>>>(Wp, Bv, bproj, ws);
    gemv_kernel<<<1024, 256, 0, stream>>>(x, ws, u);           // 8192 waves
    scan_kernel<<<2, 32, 0, stream>>>(A, C, Dm, u, (float*)d_out);
}